// LocalAutoCorr2D_20194936226266
// MI455X (gfx1250) — compile-verified
//
#include <hip/hip_runtime.h>
#include <stdint.h>

// LocalAutoCorr2D for x[4,128,128,128] f32, k=(8,8), stride=(4,4)
// out[b,iy,ix,c,dy,dx] = sum_{u,v<8} x[b,c,4iy+u,4ix+v]*xpad[b,c,4iy+u+dy,4ix+v+dx]
// (xpad = x zero-padded 4 top/left, 3 bottom/right)
//
// Roofline: ~4 GFLOP vs ~158 MB -> FP32-VALU bound (~25us) over 7us mem floor.
// Strategy: TDM async tensor DMA (global->LDS, iterate over 8 channels, HW pad
// feature creates the 136-dword padded pitch), then dense register-blocked FMA
// (64 accumulators/thread, 0.25 ds-dword/FMA via ds_load_b128).

#define HH 128
#define WW 128
#define CC 128
#define BB 4
#define NH 31
#define NW 31

#define TROWS 15
#define TCOLS 136                       // 4 halo + 128 + (3 halo + 1 align)
#define CH_PER_WG 8
#define TILE_DW (TROWS * TCOLS)         // 2040 dwords per channel
#define LDS_DW (CH_PER_WG * TILE_DW)    // 16320 dwords = 65280 B

typedef unsigned int u32x4 __attribute__((ext_vector_type(4)));
typedef int i32x8 __attribute__((ext_vector_type(8)));
typedef int i32x4 __attribute__((ext_vector_type(4)));

__global__ __launch_bounds__(256) void LocalAutoCorr2D_kernel(
    const float* __restrict__ x, float* __restrict__ out) {
  __shared__ __align__(16) float tile[LDS_DW];

  const int iy = blockIdx.x;   // 0..30
  const int cg = blockIdx.y;   // 0..15  (channel group of 8)
  const int b  = blockIdx.z;   // 0..3
  const int tid = (int)threadIdx.x;
  const int c_local = tid >> 5;       // 0..7
  const int ix = tid & 31;            // 0..31 (slot 31 is a masked dup)

  // ---- 1) zero-fill LDS (halo columns + any clamped rows stay 0) ----
  for (int i = tid; i < LDS_DW; i += 256) tile[i] = 0.0f;
  __syncthreads();

  // ---- 2) TDM DMA: 8 channels x nrows x 128 f32, global -> padded LDS ----
  const int r0 = 4 * iy - 4;                       // first (virtual) strip row
  const int row_start = r0 < 0 ? 0 : r0;           // clamp top
  int row_end = 4 * iy + 11; if (row_end > HH) row_end = HH;
  const int nrows = row_end - row_start;           // 11..15
  const int row_off = row_start - r0;              // LDS row where data begins

  if (tid < 32) {                                  // wave 0 issues the DMA
    const int c0 = cg * CH_PER_WG;
    const uint64_t gaddr = (uint64_t)(uintptr_t)
        (x + ((((size_t)b * CC + c0) * HH + row_start) * WW));
    const uint32_t lds_base = (uint32_t)(uintptr_t)(&tile[0]);
    const uint32_t lds_addr = lds_base + (uint32_t)((row_off * TCOLS + 4) * 4);

    // ---- D# group 0: count=1, lds_addr, global_addr, type=2 ----
    u32x4 g0;
    g0[0] = 1u;                                      // count=1, user mode
    g0[1] = lds_addr;                                // LDS byte address
    g0[2] = (uint32_t)(gaddr & 0xffffffffu);         // global_addr[31:0]
    g0[3] = (uint32_t)((gaddr >> 32) & 0x1ffffffu)   // global_addr[56:32]
          | (2u << 30);                              // type=2 ("image")

    // ---- D# group 1 ----
    // data_size=4B(2), iterate_enable, pad_enable,
    // pad_interval code 6 (=128 dwords), pad_amount code 7 (=8 dwords)
    // -> each 128-dword row is followed by an 8-dword LDS skip = 136 pitch,
    //    which also preserves the pre-zeroed 4+4 halo columns.
    i32x8 g1;
    g1[0] = (int)((2u << 16) | (1u << 19) | (1u << 20) | (6u << 22) | (7u << 25));
    g1[1] = (int)(((uint32_t)WW & 0xffffu) << 16);             // tensor_dim0 lo16
    g1[2] = (int)(((uint32_t)WW >> 16)
          | (((uint32_t)nrows & 0xffffu) << 16));              // tdim0 hi | tensor_dim1 lo
    g1[3] = (int)(((uint32_t)nrows >> 16) | ((uint32_t)WW << 16)); // tdim1 hi | tile_dim0=128
    g1[4] = nrows;                                             // tile_dim1 (tile_dim2=0)
    g1[5] = WW;                                                // tensor_dim0_stride lo32 (elems)
    g1[6] = (int)((((uint32_t)(HH * WW)) & 0xffffu) << 16);    // stride0 hi=0 | stride1 lo16
    g1[7] = (int)(((uint32_t)(HH * WW)) >> 16);                // stride1 hi

    // ---- D# group 2 (iterate mode): per-channel increments ----
    i32x4 g2;
    g2[0] = 0;                 // tensor_dim2 (unused, 2D tile)
    g2[1] = TILE_DW;           // lds_addr_increment  = 2040 elems (fixed pitch)
    g2[2] = HH * WW;           // global_addr_increment = 16384 elems (chan stride)
    g2[3] = (int)(7u << 16);   // gai[47:32]=0 | iterate_count=7 -> 8 channels

    i32x4 g3; g3[0] = 0; g3[1] = 0; g3[2] = 0; g3[3] = 0;

    // clang-23 / therock-10.0 form: extra (reserved/extended) int32x8 group,
    // zero-filled, then cpol.
    i32x8 g4;
    g4[0] = 0; g4[1] = 0; g4[2] = 0; g4[3] = 0;
    g4[4] = 0; g4[5] = 0; g4[6] = 0; g4[7] = 0;

    __builtin_amdgcn_tensor_load_to_lds(g0, g1, g2, g3, g4, 0);
    __builtin_amdgcn_s_wait_tensorcnt(0);
  }
  __syncthreads();

  // ---- 3) register-blocked autocorrelation ----
  // tile_c[r][t] = x[b,c, 4iy-4+r, t-4] (0 outside), r in [0,15), t in [0,136)
  //   a term:  tile[u+4][4ix+4+v]
  //   g term:  tile[u+dy][4ix+dx+v]
  const int ixc = (ix < 31) ? ix : 30;
  const float* tc = &tile[c_local * TILE_DW];

  float acc[8][8];
#pragma unroll
  for (int dy = 0; dy < 8; ++dy)
#pragma unroll
    for (int dx = 0; dx < 8; ++dx) acc[dy][dx] = 0.0f;

  for (int u = 0; u < 8; ++u) {  // dynamic: keeps code size ~512 FMA/iter
    const float4 A0 = *(const float4*)(tc + (u + 4) * TCOLS + 4 * ixc + 4);
    const float4 A1 = *(const float4*)(tc + (u + 4) * TCOLS + 4 * ixc + 8);
    const float a[8] = {A0.x, A0.y, A0.z, A0.w, A1.x, A1.y, A1.z, A1.w};
#pragma unroll
    for (int dy = 0; dy < 8; ++dy) {
      const float* gr = tc + (u + dy) * TCOLS + 4 * ixc;
      const float4 G0 = *(const float4*)(gr + 0);
      const float4 G1 = *(const float4*)(gr + 4);
      const float4 G2 = *(const float4*)(gr + 8);
      const float4 G3 = *(const float4*)(gr + 12);
      const float g[16] = {G0.x, G0.y, G0.z, G0.w, G1.x, G1.y, G1.z, G1.w,
                           G2.x, G2.y, G2.z, G2.w, G3.x, G3.y, G3.z, G3.w};
#pragma unroll
      for (int dx = 0; dx < 8; ++dx)
#pragma unroll
        for (int v = 0; v < 8; ++v)
          acc[dy][dx] = __builtin_fmaf(a[v], g[dx + v], acc[dy][dx]);
    }
  }

  // ---- 4) store: 64 contiguous floats per thread (dense in aggregate) ----
  if (ix < 31) {
    const int c = cg * CH_PER_WG + c_local;
    const size_t base = ((((size_t)b * NH + iy) * NW + ix) * CC + c) * 64;
    float4* o = (float4*)(out + base);
#pragma unroll
    for (int dy = 0; dy < 8; ++dy) {
      o[2 * dy + 0] = make_float4(acc[dy][0], acc[dy][1], acc[dy][2], acc[dy][3]);
      o[2 * dy + 1] = make_float4(acc[dy][4], acc[dy][5], acc[dy][6], acc[dy][7]);
    }
  }
}

extern "C" void kernel_launch(void* const* d_in, const int* in_sizes, int n_in,
                              void* d_out, int out_size, void* d_ws, size_t ws_size,
                              hipStream_t stream) {
  (void)in_sizes; (void)n_in; (void)d_ws; (void)ws_size; (void)out_size;
  const float* x = (const float*)d_in[0];
  float* out = (float*)d_out;
  dim3 grid(NH, CC / CH_PER_WG, BB);
  LocalAutoCorr2D_kernel<<<grid, 256, 0, stream>>>(x, out);
}